// KabschGDT_TS_3582002724943
// MI455X (gfx1250) — compile-verified
//
#include <hip/hip_runtime.h>

typedef __attribute__((ext_vector_type(2))) float v2f;
typedef __attribute__((ext_vector_type(8))) float v8f;

#if defined(__has_builtin)
#  if __has_builtin(__builtin_amdgcn_wmma_f32_16x16x4_f32)
#    define HAVE_WMMA_F32X4 1
#  endif
#endif

#define THREADS 256

// ---------------------------------------------------------------------------
// Pass 1: per-batch raw sums (Sigma p_i q_j, Sigma p, Sigma q) + Kabsch
// rotation via Horn's quaternion method (top eigenvector of 4x4 K).
// One block per batch, 256 threads, fully coalesced streaming.
// Writes 12 floats per batch to ws: R (row-major, 9) + t (3).
// ---------------------------------------------------------------------------
__global__ void cov_kabsch_kernel(const float* __restrict__ P,
                                  const float* __restrict__ Q,
                                  float* __restrict__ rt, int N) {
  const int b = blockIdx.x;
  const int tid = threadIdx.x;
  const float* Pb = P + (size_t)b * N * 3;
  const float* Qb = Q + (size_t)b * N * 3;

  float s[15];
#pragma unroll
  for (int k = 0; k < 15; ++k) s[k] = 0.0f;

#pragma unroll 4
  for (int i = tid; i < N; i += THREADS) {
    const float px = Pb[3 * i + 0], py = Pb[3 * i + 1], pz = Pb[3 * i + 2];
    const float qx = Qb[3 * i + 0], qy = Qb[3 * i + 1], qz = Qb[3 * i + 2];
    s[0] = fmaf(px, qx, s[0]);  s[1] = fmaf(px, qy, s[1]);  s[2] = fmaf(px, qz, s[2]);
    s[3] = fmaf(py, qx, s[3]);  s[4] = fmaf(py, qy, s[4]);  s[5] = fmaf(py, qz, s[5]);
    s[6] = fmaf(pz, qx, s[6]);  s[7] = fmaf(pz, qy, s[7]);  s[8] = fmaf(pz, qz, s[8]);
    s[9] += px;  s[10] += py;  s[11] += pz;
    s[12] += qx; s[13] += qy;  s[14] += qz;
  }

  __shared__ float red[THREADS * 15];
#pragma unroll
  for (int k = 0; k < 15; ++k) red[tid * 15 + k] = s[k];
  __syncthreads();
  for (int stride = THREADS / 2; stride > 0; stride >>= 1) {
    if (tid < stride) {
#pragma unroll
      for (int k = 0; k < 15; ++k)
        red[tid * 15 + k] += red[(tid + stride) * 15 + k];
    }
    __syncthreads();
  }

  if (tid == 0) {
    const float inv = 1.0f / (float)N;
    const float spx = red[9],  spy = red[10], spz = red[11];
    const float sqx = red[12], sqy = red[13], sqz = red[14];
    // Centered cross-covariance A_ij = Sum p_i q_j - (Sum p_i)(Sum q_j)/N
    const float Axx = red[0] - spx * sqx * inv, Axy = red[1] - spx * sqy * inv, Axz = red[2] - spx * sqz * inv;
    const float Ayx = red[3] - spy * sqx * inv, Ayy = red[4] - spy * sqy * inv, Ayz = red[5] - spy * sqz * inv;
    const float Azx = red[6] - spz * sqx * inv, Azy = red[7] - spz * sqy * inv, Azz = red[8] - spz * sqz * inv;

    // Horn's 4x4 key matrix (order w,x,y,z); its top eigenvector is the
    // quaternion of the optimal proper rotation (== Kabsch with det fix).
    const float K00 =  Axx + Ayy + Azz;
    const float K11 =  Axx - Ayy - Azz;
    const float K22 = -Axx + Ayy - Azz;
    const float K33 = -Axx - Ayy + Azz;
    const float K01 = Azy - Ayz;
    const float K02 = Axz - Azx;
    const float K03 = Ayx - Axy;
    const float K12 = Axy + Ayx;
    const float K13 = Axz + Azx;
    const float K23 = Ayz + Azy;

    // Gershgorin shift -> K + sh*I is PSD, power iteration finds lambda_max.
    float sh = fabsf(K00) + fabsf(K01) + fabsf(K02) + fabsf(K03);
    sh = fmaxf(sh, fabsf(K01) + fabsf(K11) + fabsf(K12) + fabsf(K13));
    sh = fmaxf(sh, fabsf(K02) + fabsf(K12) + fabsf(K22) + fabsf(K23));
    sh = fmaxf(sh, fabsf(K03) + fabsf(K13) + fabsf(K23) + fabsf(K33));

    float w = 1.0f, x = 0.23f, y = -0.37f, z = 0.41f;  // generic start
#pragma unroll 1
    for (int it = 0; it < 96; ++it) {
      const float nw = K00 * w + K01 * x + K02 * y + K03 * z + sh * w;
      const float nx = K01 * w + K11 * x + K12 * y + K13 * z + sh * x;
      const float ny = K02 * w + K12 * x + K22 * y + K23 * z + sh * y;
      const float nz = K03 * w + K13 * x + K23 * y + K33 * z + sh * z;
      const float r = rsqrtf(nw * nw + nx * nx + ny * ny + nz * nz + 1e-30f);
      w = nw * r; x = nx * r; y = ny * r; z = nz * r;
    }

    // Quaternion -> rotation matrix
    const float R00 = 1.0f - 2.0f * (y * y + z * z);
    const float R01 = 2.0f * (x * y - w * z);
    const float R02 = 2.0f * (x * z + w * y);
    const float R10 = 2.0f * (x * y + w * z);
    const float R11 = 1.0f - 2.0f * (x * x + z * z);
    const float R12 = 2.0f * (y * z - w * x);
    const float R20 = 2.0f * (x * z - w * y);
    const float R21 = 2.0f * (y * z + w * x);
    const float R22 = 1.0f - 2.0f * (x * x + y * y);

    const float pmx = spx * inv, pmy = spy * inv, pmz = spz * inv;
    const float cmx = sqx * inv, cmy = sqy * inv, cmz = sqz * inv;
    const float tx = pmx - (R00 * cmx + R01 * cmy + R02 * cmz);
    const float ty = pmy - (R10 * cmx + R11 * cmy + R12 * cmz);
    const float tz = pmz - (R20 * cmx + R21 * cmy + R22 * cmz);

    float* o = rt + (size_t)b * 12;
    o[0] = R00; o[1] = R01; o[2] = R02;
    o[3] = R10; o[4] = R11; o[5] = R12;
    o[6] = R20; o[7] = R21; o[8] = R22;
    o[9] = tx;  o[10] = ty; o[11] = tz;
  }
}

// ---------------------------------------------------------------------------
// Pass 2: aligned = R*q + t via V_WMMA_F32_16X16X4_F32.
//   A(16x4): rows 0..2 = [R | t]    (M = output component, K = input comp + 1)
//   B(4x16): col j = (qx,qy,qz,1) of point j
//   D: VGPR0/1/2, lanes 0..15 = aligned x/y/z of the 16 points (SoA in lanes)
// Fully branchless loop body (no exec-mask churn), x2 unrolled: 2 WMMAs and
// ~10 independent global loads in flight per iteration.
// ---------------------------------------------------------------------------
__global__ void gdt_kernel(const float* __restrict__ P,
                           const float* __restrict__ Q,
                           const float* __restrict__ rt,
                           float* __restrict__ out, int N) {
  const int b = blockIdx.x;
  const int tid = threadIdx.x;
  const int lane = tid & 31;
  const int wave = tid >> 5;
  const int nwaves = THREADS >> 5;
  const int m = lane & 15;     // M row (A) / N col (B) index
  const int half = lane >> 4;  // K half: 0 -> K={0,1}, 1 -> K={2,3}
  const int co = half * 2;     // component offset for K-lane pair

  const float* Pb = P + (size_t)b * N * 3;
  const float* Qb = Q + (size_t)b * N * 3;
  const float* o = rt + (size_t)b * 12;

  float Rm[9], tv[3];
#pragma unroll
  for (int k = 0; k < 9; ++k) Rm[k] = o[k];
#pragma unroll
  for (int k = 0; k < 3; ++k) tv[k] = o[9 + k];

#if HAVE_WMMA_F32X4
  // A operand is loop-invariant: lane l holds A(M=l%16, K=2*(l/16)+{0,1})
  v2f av;
  {
    const bool row = (m < 3);
    const int mi = row ? m : 0;
    const float a0 = Rm[mi * 3 + co];                    // R[m][0] or R[m][2]
    const float a1 = half ? tv[mi] : Rm[mi * 3 + 1];     // t[m]   or R[m][1]
    av.x = row ? a0 : 0.0f;
    av.y = row ? a1 : 0.0f;
  }
#endif

  int cnt = 0;
  for (int base = wave * 32; base < N; base += nwaves * 32) {
    const int idx0 = base + m;
    const int idx1 = base + 16 + m;
    const bool inb0 = idx0 < N;
    const bool inb1 = idx1 < N;
    const int ci0 = inb0 ? idx0 : (N - 1);
    const int ci1 = inb1 ? idx1 : (N - 1);

    // ---- all global loads issued up front, branchless --------------------
    const float q0a = Qb[3 * ci0 + co];   // qx (half 0) or qz (half 1)
    const float q0b = Qb[3 * ci0 + 1];    // qy (same cacheline, no extra traffic)
    const float q1a = Qb[3 * ci1 + co];
    const float q1b = Qb[3 * ci1 + 1];
    const float p0x = Pb[3 * ci0 + 0];    // lanes 16..31 re-request lanes 0..15's
    const float p0y = Pb[3 * ci0 + 1];    // addresses -> coalesced, no extra HBM
    const float p0z = Pb[3 * ci0 + 2];
    const float p1x = Pb[3 * ci1 + 0];
    const float p1y = Pb[3 * ci1 + 1];
    const float p1z = Pb[3 * ci1 + 2];

#if HAVE_WMMA_F32X4
    v2f bv0, bv1;
    bv0.x = q0a; bv0.y = half ? 1.0f : q0b;
    bv1.x = q1a; bv1.y = half ? 1.0f : q1b;

    v8f c0 = {}, c1 = {};
    c0 = __builtin_amdgcn_wmma_f32_16x16x4_f32(
        false, av, false, bv0, (short)0, c0, false, false);
    c1 = __builtin_amdgcn_wmma_f32_16x16x4_f32(
        false, av, false, bv1, (short)0, c1, false, false);
    const float a0x = c0[0], a0y = c0[1], a0z = c0[2];
    const float a1x = c1[0], a1y = c1[1], a1z = c1[2];
#else
    // VALU fallback (no WMMA builtin available); q*a/q*b hold x/y for half==0
    const float q0z = Qb[3 * ci0 + 2];
    const float q1z = Qb[3 * ci1 + 2];
    const float a0x = fmaf(Rm[0], q0a, fmaf(Rm[1], q0b, fmaf(Rm[2], q0z, tv[0])));
    const float a0y = fmaf(Rm[3], q0a, fmaf(Rm[4], q0b, fmaf(Rm[5], q0z, tv[1])));
    const float a0z = fmaf(Rm[6], q0a, fmaf(Rm[7], q0b, fmaf(Rm[8], q0z, tv[2])));
    const float a1x = fmaf(Rm[0], q1a, fmaf(Rm[1], q1b, fmaf(Rm[2], q1z, tv[0])));
    const float a1y = fmaf(Rm[3], q1a, fmaf(Rm[4], q1b, fmaf(Rm[5], q1z, tv[1])));
    const float a1z = fmaf(Rm[6], q1a, fmaf(Rm[7], q1b, fmaf(Rm[8], q1z, tv[2])));
#endif

    // ---- distances + threshold hits, arithmetically masked ---------------
    {
      const float dx = p0x - a0x, dy = p0y - a0y, dz = p0z - a0z;
      const float d2 = dx * dx + dy * dy + dz * dz;
      const int hits = (d2 < 1.0f) + (d2 < 4.0f) + (d2 < 16.0f) + (d2 < 64.0f);
      cnt += (half == 0 && inb0) ? hits : 0;
    }
    {
      const float dx = p1x - a1x, dy = p1y - a1y, dz = p1z - a1z;
      const float d2 = dx * dx + dy * dy + dz * dz;
      const int hits = (d2 < 1.0f) + (d2 < 4.0f) + (d2 < 16.0f) + (d2 < 64.0f);
      cnt += (half == 0 && inb1) ? hits : 0;
    }
  }

  __shared__ int ired[THREADS];
  ired[tid] = cnt;
  __syncthreads();
  for (int stride = THREADS / 2; stride > 0; stride >>= 1) {
    if (tid < stride) ired[tid] += ired[tid + stride];
    __syncthreads();
  }
  if (tid == 0) out[b] = (float)ired[0] / (4.0f * (float)N);
}

// ---------------------------------------------------------------------------
extern "C" void kernel_launch(void* const* d_in, const int* in_sizes, int n_in,
                              void* d_out, int out_size, void* d_ws, size_t ws_size,
                              hipStream_t stream) {
  const float* P = (const float*)d_in[0];  // coords_pred [B,N,3]
  const float* Q = (const float*)d_in[1];  // coords      [B,N,3]
  float* out = (float*)d_out;              // gdt_ts [B]
  float* rt = (float*)d_ws;                // per-batch R(9)+t(3)

  const int B = out_size;
  const int N = in_sizes[0] / (3 * B);

  cov_kabsch_kernel<<<B, THREADS, 0, stream>>>(P, Q, rt, N);
  gdt_kernel<<<B, THREADS, 0, stream>>>(P, Q, rt, out, N);
}